// NeuralNetwork_62397284876811
// MI455X (gfx1250) — compile-verified
//
#include <hip/hip_runtime.h>
#include <math.h>

#define I_  128
#define L_  8
#define T_  128
#define S_  128
#define SP1 129
#define SLAB (S_ * SP1)      // 16512 floats = 66048 bytes per qkv matrix
#define EPSV 1e-5f

typedef float v2f __attribute__((ext_vector_type(2)));
typedef float v8f __attribute__((ext_vector_type(8)));
typedef unsigned int u32x4 __attribute__((ext_vector_type(4)));
typedef int i32x4 __attribute__((ext_vector_type(4)));
typedef int i32x8 __attribute__((ext_vector_type(8)));

__device__ __forceinline__ float gelu_tanh(float g) {
    const float c = 0.7978845608028654f; // sqrt(2/pi)
    return 0.5f * g * (1.0f + tanhf(c * (g + 0.044715f * g * g * g)));
}

// ---------------------------------------------------------------------------
// TDM: DMA one contiguous 16512-float slab from global memory into LDS.
// D# per CDNA5 ISA:
//   group0: count=1 | lds_addr | global_addr[56:0] | type=2
//   group1: data_size=4B, tensor_dim0=tile_dim0=16512, tensor/tile_dim1=1,
//           tensor_dim0_stride=16512, no pad/iterate/multicast
// Groups 2/3 zero (tile dims 2+ unused). Tracked by TENSORcnt.
// (6-arg builtin form: g0, g1, g2, g3, extra, cpol)
// ---------------------------------------------------------------------------
__device__ __forceinline__ void tdm_load_slab(const float* gsrc, unsigned lds_off) {
    const unsigned long long ga = (unsigned long long)(uintptr_t)gsrc;
    u32x4 g0;
    g0[0] = 1u;                                               // count=1, no gather
    g0[1] = lds_off;                                          // LDS byte address
    g0[2] = (unsigned)(ga & 0xFFFFFFFFull);                   // global_addr[31:0]
    g0[3] = ((unsigned)(ga >> 32) & 0x01FFFFFFu) | 0x80000000u; // [56:32] | type=2
    i32x8 g1;
    g1[0] = 0x00020000;       // workgroup_mask=0, data_size=2 (4 bytes)
    g1[1] = 0x40800000;       // tensor_dim0 = 16512 (low 16 bits at [31:16])
    g1[2] = 0x00010000;       // tensor_dim0 hi=0, tensor_dim1 = 1
    g1[3] = 0x40800000;       // tensor_dim1 hi=0, tile_dim0 = 16512
    g1[4] = 0x00000001;       // tile_dim1 = 1, tile_dim2 = 0
    g1[5] = SLAB;             // tensor_dim0_stride = 16512
    g1[6] = 0x40800000;       // tensor_dim1_stride low16 = 16512
    g1[7] = 0;                // tensor_dim1_stride hi
    i32x4 z4 = {0, 0, 0, 0};
    i32x8 z8 = {0, 0, 0, 0, 0, 0, 0, 0};
    __builtin_amdgcn_tensor_load_to_lds(g0, g1, z4, z4, z8, 0);
}

// ---------------------------------------------------------------------------
// Per-layer prologue: 1 block, 128 threads.
// LayerNorm + affine + token attention (attn_t[b], amask=0) -> cur[0..127]
// ---------------------------------------------------------------------------
__global__ void __launch_bounds__(128)
layer_pre(const float* __restrict__ x,
          const float* __restrict__ attn_t,
          const float* __restrict__ norm_params,
          float* __restrict__ values,
          float* __restrict__ cur, int b) {
    __shared__ float red[S_];
    __shared__ float sv[S_], sq[S_], sk[S_], svv[S_];
    const int tid = threadIdx.x;

    if (b == 0) values[tid] = x[tid];
    __syncthreads();

    float v = values[b * S_ + tid];

    red[tid] = v; __syncthreads();
    for (int off = 64; off > 0; off >>= 1) { if (tid < off) red[tid] += red[tid + off]; __syncthreads(); }
    const float mu = red[0] * (1.0f / S_);
    __syncthreads();
    red[tid] = v * v; __syncthreads();
    for (int off = 64; off > 0; off >>= 1) { if (tid < off) red[tid] += red[tid + off]; __syncthreads(); }
    const float var = red[0] * (1.0f / S_) - mu * mu;
    __syncthreads();

    v = (v - mu) * rsqrtf(var + EPSV);
    v = v * norm_params[(b * 2 + 0) * S_ + tid] + norm_params[(b * 2 + 1) * S_ + tid];
    sv[tid] = v;
    __syncthreads();

    const float* At = attn_t + (size_t)b * 3 * SLAB;
    for (int m = 0; m < 3; ++m) {
        const float* Wm = At + (size_t)m * SLAB + (size_t)tid * SP1;
        float acc = Wm[S_];                      // bias = last column
        #pragma unroll 4
        for (int c = 0; c < S_; ++c) acc += Wm[c] * sv[c];
        (m == 0 ? sq : (m == 1 ? sk : svv))[tid] = acc;
    }
    __syncthreads();

    const float scale = rsqrtf((float)S_);
    const float qi = sq[tid];
    float mx = -3.4e38f;
    for (int j = 0; j < S_; ++j) mx = fmaxf(mx, qi * sk[j] * scale);
    float sum = 0.f, o = 0.f;
    for (int j = 0; j < S_; ++j) {
        float e = __expf(qi * sk[j] * scale - mx);
        sum += e; o += e * svv[j];
    }
    cur[tid] = o / sum + v;
}

// ---------------------------------------------------------------------------
// Per-layer main: 128 blocks (one per t), 256 threads = 8 waves.
// TDM double-buffers the three 66KB qkv slabs into LDS while WMMA consumes
// them; mask streamed as float4; softmax + @v + residual; masked dot; gelu.
// ---------------------------------------------------------------------------
__global__ void __launch_bounds__(256)
layer_main(const float* __restrict__ attn_n,
           const float* __restrict__ attn_mask_n,
           const float* __restrict__ Wp,
           const float* __restrict__ maskp,
           const float* __restrict__ ada,
           const float* __restrict__ cur,
           float* __restrict__ values,
           float* __restrict__ d_out, int b) {
    __shared__ float buf[2][SLAB];               // 2 x 66048B double buffer
    __shared__ float sv[S_], sq[S_], sk[S_], svv[S_];
    __shared__ float sred[256];

    const int tid = threadIdx.x;
    const int t   = blockIdx.x;
    const size_t bt = (size_t)b * T_ + t;

    const float* slabsrc = attn_n + bt * 3 * (size_t)SLAB;
    const unsigned lds0 = (unsigned)(uintptr_t)&buf[0][0];
    const unsigned lds1 = (unsigned)(uintptr_t)&buf[1][0];

    // wave 0 kicks off the first DMA immediately
    if (tid < 32) tdm_load_slab(slabsrc, lds0);

    if (tid < S_) sv[tid] = cur[tid];

    const int lane    = tid & 31;
    const int wv      = tid >> 5;
    const int n0      = wv * 16;
    const int halfsel = (lane >= 16) ? 2 : 0;
    const int nn      = n0 + (lane & 15);
    const float scale = rsqrtf((float)S_);

    #pragma unroll
    for (int m = 0; m < 3; ++m) {
        if (tid < 32) {
            if (m == 0) {            // prefetch slab 1 into buf1, wait slab 0
                tdm_load_slab(slabsrc + (size_t)SLAB, lds1);
                __builtin_amdgcn_s_wait_tensorcnt(1);
            } else if (m == 1) {     // prefetch slab 2 into buf0, wait slab 1
                tdm_load_slab(slabsrc + 2 * (size_t)SLAB, lds0);
                __builtin_amdgcn_s_wait_tensorcnt(1);
            } else {                 // wait slab 2
                __builtin_amdgcn_s_wait_tensorcnt(0);
            }
        }
        __syncthreads();

        const float* slab = buf[m & 1];
        // y[n0+n] = sum_k W[n0+n,k] * x[k] via WMMA f32 16x16x4
        // B[k,n] = W[n0+n, 4*kc+k] (weights transposed into B)
        // A[r,k] = x[4*kc+k]       (input broadcast over A rows)
        v8f acc = {};
        #pragma unroll
        for (int kc = 0; kc < 32; ++kc) {
            const int kb = kc * 4 + halfsel;
            v2f a, bm;
            a[0]  = sv[kb];
            a[1]  = sv[kb + 1];
            bm[0] = slab[nn * SP1 + kb];
            bm[1] = slab[nn * SP1 + kb + 1];
            acc = __builtin_amdgcn_wmma_f32_16x16x4_f32(
                false, a, false, bm, (short)0, acc, false, false);
        }
        if (lane < 16) {                          // D row M=0, N=lane
            float y = acc[0] + slab[nn * SP1 + S_];   // + bias column
            (m == 0 ? sq : (m == 1 ? sk : svv))[nn] = y;
        }
        __syncthreads();
    }

    // ---- scores = outer(q,k)*scale - mask  -> buf[1] (slab 1 is consumed) --
    float* sc = buf[1];
    const float4* mk = reinterpret_cast<const float4*>(
        attn_mask_n + bt * (size_t)(S_ * S_));
    for (int i4 = tid; i4 < (S_ * S_) / 4; i4 += 256) {
        float4 mv = mk[i4];
        const int e0  = i4 * 4;
        const int row = e0 >> 7;
        const int col = e0 & (S_ - 1);
        const float qr = sq[row];
        float4 s4;
        s4.x = qr * sk[col + 0] * scale - mv.x;
        s4.y = qr * sk[col + 1] * scale - mv.y;
        s4.z = qr * sk[col + 2] * scale - mv.z;
        s4.w = qr * sk[col + 3] * scale - mv.w;
        reinterpret_cast<float4*>(sc)[i4] = s4;
    }
    __syncthreads();

    // ---- row softmax + @v + residual -> vals_n (stored into sq) ----
    float vn = 0.f;
    if (tid < S_) {
        float mx = -3.4e38f;
        for (int j = 0; j < S_; ++j) mx = fmaxf(mx, sc[tid * S_ + j]);
        float sum = 0.f, o = 0.f;
        for (int j = 0; j < S_; ++j) {
            float e = __expf(sc[tid * S_ + j] - mx);
            sum += e; o += e * svv[j];
        }
        vn = o / sum + sv[tid];
    }
    __syncthreads();
    if (tid < S_) sq[tid] = vn;
    __syncthreads();

    // ---- aff = sum_s W[b,t,s]*mask[b,t,s]*vals_n[s] + bias ----
    float p = 0.f;
    if (tid < S_)
        p = Wp[bt * SP1 + tid] * maskp[bt * S_ + tid] * sq[tid];
    sred[tid] = p; __syncthreads();
    for (int off = 128; off > 0; off >>= 1) { if (tid < off) sred[tid] += sred[tid + off]; __syncthreads(); }

    if (tid == 0) {
        const float aff = sred[0] + Wp[bt * SP1 + S_];
        if (b == L_ - 1) {
            d_out[t] = aff;
        } else {
            const float g = aff * ada[bt * 2 + 0];
            values[I_ + b * T_ + t] = gelu_tanh(g) * ada[bt * 2 + 1];
        }
    }
}

// ---------------------------------------------------------------------------
extern "C" void kernel_launch(void* const* d_in, const int* in_sizes, int n_in,
                              void* d_out, int out_size, void* d_ws, size_t ws_size,
                              hipStream_t stream) {
    const float* x           = (const float*)d_in[0];
    const float* W           = (const float*)d_in[1];
    const float* mask        = (const float*)d_in[2];
    const float* attn_t      = (const float*)d_in[3];
    const float* attn_n      = (const float*)d_in[4];
    const float* attn_mask_n = (const float*)d_in[5];
    const float* norm_params = (const float*)d_in[6];
    const float* ada         = (const float*)d_in[7];
    (void)in_sizes; (void)n_in; (void)out_size; (void)ws_size;

    float* values = (float*)d_ws;            // I + L*T = 1152 floats
    float* cur    = values + (I_ + L_ * T_); // 128 floats

    for (int b = 0; b < L_; ++b) {
        layer_pre<<<1, 128, 0, stream>>>(x, attn_t, norm_params, values, cur, b);
        layer_main<<<T_, 256, 0, stream>>>(attn_n, attn_mask_n, W, mask, ada,
                                           cur, values, (float*)d_out, b);
    }
}